// TimeVariationAdaptiveRecursiveFusion_39779987095794
// MI455X (gfx1250) — compile-verified
//
#include <hip/hip_runtime.h>
#include <math.h>

// Problem dims
#define S_ 4
#define K_ 4
#define B_ 8
#define C_ 64
#define H_ 96
#define W_ 96
#define DIM_ 64
#define N_ 16
#define HW_ (H_*W_)          // 9216
#define CHW_ (C_*HW_)        // 589824
#define KSTRIDE_ (B_*CHW_)   // 4718592 (stride of k within a stage)

// ---------------------------------------------------------------------------
// Kernel 1: per-cell pooled stats (mean_c, var_c, grad-strength) + per-(s,k)
// global max of per-pixel gstr (for alpha normalization).
// grid = S*K*B*16 cells, 256 threads.
// ---------------------------------------------------------------------------
__global__ void k_stats(const float* __restrict__ T,
                        float* __restrict__ pooled,   // [S,K,B,3,16]
                        float* __restrict__ gmax) {   // [S,K]
  int bid = blockIdx.x;
  int cell = bid & 15, b = (bid >> 4) & 7, k = (bid >> 7) & 3, s = bid >> 9;
  int gy = cell >> 2, gx = cell & 3;
  int tid = threadIdx.x;
  const float* Tb = T + ((s*K_ + k)*B_ + b) * CHW_;
  float am = 0.f, av = 0.f, ag = 0.f, mg = 0.f;
  for (int p = tid; p < 576; p += 256) {
    int py = p / 24, px = p - py * 24;
    int h = gy * 24 + py, w = gx * 24 + px;
    const float* p0 = Tb + h * W_ + w;
    float sxs = (w < W_-1) ? 1.f : -1.f; int ox = (w < W_-1) ?  1 :  -1;
    float sys = (h < H_-1) ? 1.f : -1.f; int oy = (h < H_-1) ? W_ : -W_;
    float s1 = 0.f, s2 = 0.f, g = 0.f;
    for (int c = 0; c < C_; ++c) {
      float tc = p0[0];
      float dx = sxs * (p0[ox] - tc);
      float dy = sys * (p0[oy] - tc);
      s1 += tc; s2 += tc * tc;
      g += sqrtf(dx*dx + dy*dy + 1e-8f);
      p0 += HW_;
    }
    float pm = s1 * (1.f/64.f);
    float pv = (s2 - s1*s1*(1.f/64.f)) * (1.f/63.f);  // ddof=1
    float pg = g * (1.f/64.f);
    am += pm; av += pv; ag += pg; mg = fmaxf(mg, pg);
  }
  __shared__ float rs[1024];
  rs[tid] = am; rs[256+tid] = av; rs[512+tid] = ag; rs[768+tid] = mg;
  for (int off = 128; off > 0; off >>= 1) {
    __syncthreads();
    if (tid < off) {
      rs[tid]     += rs[tid+off];
      rs[256+tid] += rs[256+tid+off];
      rs[512+tid] += rs[512+tid+off];
      rs[768+tid]  = fmaxf(rs[768+tid], rs[768+tid+off]);
    }
  }
  if (tid == 0) {
    int base = ((s*K_ + k)*B_ + b) * 48;
    int n = gy*4 + gx;
    pooled[base +       n] = rs[0]   * (1.f/576.f);
    pooled[base + 16 +  n] = rs[256] * (1.f/576.f);
    pooled[base + 32 +  n] = rs[512] * (1.f/576.f);
    atomicMax((int*)(gmax + s*K_ + k), __float_as_int(rs[768]));  // gstr > 0
  }
}

// ---------------------------------------------------------------------------
// Kernel 2: tiny SE-MLP producing z[S,K,B,N,64] and alpha[S,K,B,N].
// grid = S*K*B, 64 threads.
// ---------------------------------------------------------------------------
__global__ void k_se(const float* __restrict__ pooled, const float* __restrict__ gmax,
                     const float* __restrict__ W1, const float* __restrict__ b1,
                     const float* __restrict__ W2, const float* __restrict__ b2,
                     float* __restrict__ z, float* __restrict__ alpha) {
  int bid = blockIdx.x;
  int b = bid & 7, k = (bid >> 3) & 3, s = bid >> 5;
  int d = threadIdx.x;
  __shared__ float hs[64 * 16];
  int pbase = ((s*K_ + k)*B_ + b) * 48;
  float w0 = W1[(s*64 + d)*3 + 0], w1v = W1[(s*64 + d)*3 + 1],
        w2v = W1[(s*64 + d)*3 + 2], bb = b1[s*64 + d];
  for (int n = 0; n < 16; ++n) {
    float v = w0*pooled[pbase + n] + w1v*pooled[pbase + 16 + n] +
              w2v*pooled[pbase + 32 + n] + bb;
    hs[d*16 + n] = fmaxf(v, 0.f);
  }
  __syncthreads();
  const float* W2r = W2 + (s*64 + d) * 64;
  float bb2 = b2[s*64 + d];
  int zbase = ((s*K_ + k)*B_ + b) * 16 * 64;
  for (int n = 0; n < 16; ++n) {
    float acc = bb2;
    for (int c = 0; c < 64; ++c) acc += W2r[c] * hs[c*16 + n];
    z[zbase + n*64 + d] = acc;
  }
  if (d < 16)
    alpha[((s*K_ + k)*B_ + b)*16 + d] =
        pooled[pbase + 32 + d] * 0.9f / (gmax[s*K_ + k] + 1e-8f) + 0.1f;
}

// ---------------------------------------------------------------------------
// WMMA 64x64x64 fp32 GEMM on LDS matrices. 128 threads = 4 wave32; wave w owns
// output rows [16w,16w+16). Uses V_WMMA_F32_16X16X4_F32.
// A 16x4 layout: lanes 0-15 -> K+0..1 in v0..v1; lanes 16-31 -> K+2..3.
// B 4x16 layout: mirrored. C/D 16x16: VGPR r = rows r (lanes 0-15) / r+8.
// ---------------------------------------------------------------------------
typedef __attribute__((ext_vector_type(2))) float v2f;
typedef __attribute__((ext_vector_type(8))) float v8f;

enum { GEMM_STORE = 0, GEMM_SUB = 2 };

template<int MODE>
__device__ __forceinline__ void gemm64(const float* A, const float* B, float* D, int tid) {
  const int wave = tid >> 5;
  const int lane = tid & 31;
  const int half = lane >> 4;
  const int l16  = lane & 15;
  const int m0   = wave << 4;
  v8f acc0 = {0.f,0.f,0.f,0.f,0.f,0.f,0.f,0.f};
  v8f acc1 = acc0, acc2 = acc0, acc3 = acc0;
  const float* Arow = A + (m0 + l16) * 64 + (half << 1);
  const float* Bh   = B + (half << 1) * 64 + l16;
#pragma unroll
  for (int kt = 0; kt < 16; ++kt) {
    v2f a;  a.x  = Arow[0];  a.y  = Arow[1];
    v2f b0; b0.x = Bh[0];    b0.y = Bh[64];
    v2f b1; b1.x = Bh[16];   b1.y = Bh[80];
    v2f b2; b2.x = Bh[32];   b2.y = Bh[96];
    v2f b3; b3.x = Bh[48];   b3.y = Bh[112];
    acc0 = __builtin_amdgcn_wmma_f32_16x16x4_f32(false, a, false, b0, (short)0, acc0, false, false);
    acc1 = __builtin_amdgcn_wmma_f32_16x16x4_f32(false, a, false, b1, (short)0, acc1, false, false);
    acc2 = __builtin_amdgcn_wmma_f32_16x16x4_f32(false, a, false, b2, (short)0, acc2, false, false);
    acc3 = __builtin_amdgcn_wmma_f32_16x16x4_f32(false, a, false, b3, (short)0, acc3, false, false);
    Arow += 4;
    Bh   += 4 * 64;
  }
  float* Drow = D + (m0 + (half << 3)) * 64 + l16;
#pragma unroll
  for (int r = 0; r < 8; ++r) {
    if (MODE == GEMM_SUB) {
      Drow[0]  -= acc0[r]; Drow[16] -= acc1[r];
      Drow[32] -= acc2[r]; Drow[48] -= acc3[r];
    } else {
      Drow[0]  = acc0[r];  Drow[16] = acc1[r];
      Drow[32] = acc2[r];  Drow[48] = acc3[r];
    }
    Drow += 64;
  }
}

// ---------------------------------------------------------------------------
// Kernel 3: Kalman scan. One block per (s,b,n): 512 blocks x 128 threads.
// All 64x64 matrices live in (dynamic) LDS; GEMMs via WMMA fp32.
// ---------------------------------------------------------------------------
__global__ void k_kalman(const float* __restrict__ Q, const float* __restrict__ Hm,
                         const float* __restrict__ r_base, const float* __restrict__ z,
                         const float* __restrict__ alpha, float* __restrict__ x_fin) {
  extern __shared__ float smf[];
  float* Hs = smf;            // 4096  H_s
  float* P  = smf + 4096;     // 4096  P (becomes Pp in-place)
  float* HP = smf + 8192;     // 4096  H @ Pp
  float* PH = smf + 12288;    // 4096  Pp @ H, then Kk (in place, per-wave rows)
  float* Sm = smf + 16384;    // 4096  S, then S^-1 (in-place Gauss-Jordan)
  float* xv = smf + 20480;    // 64    state
  float* iv = smf + 20544;    // 64    innovation
  float* tv = smf + 20608;    // 64    pivot column scratch
  int bid = blockIdx.x;
  int n = bid & 15, b = (bid >> 4) & 7, s = bid >> 7;
  int tid = threadIdx.x;
  const float* Hg = Hm + s * 4096;
  const float* Qg = Q  + s * 4096;
  for (int i = tid; i < 4096; i += 128) {
    Hs[i] = Hg[i];
    P[i]  = (i % 65 == 0) ? 0.1f : 0.f;   // P0 = 0.1 I
  }
  if (tid < 64) xv[tid] = z[(((s*K_ + 0)*B_ + b)*16 + n)*64 + tid];
  __syncthreads();

  for (int k = 1; k < K_; ++k) {
    float ak = alpha[((s*K_ + k)*B_ + b)*16 + n];
    // Pp = P + Q
    for (int i = tid; i < 4096; i += 128) P[i] += Qg[i];
    __syncthreads();
    gemm64<GEMM_STORE>(Hs, P, HP, tid);       // HP = H @ Pp
    __syncthreads();
    gemm64<GEMM_STORE>(HP, Hs, Sm, tid);      // S = HP @ H
    __syncthreads();
    if (tid < 64) Sm[tid*65] += r_base[tid] / (ak + 1e-8f) + 1e-6f;  // + Rk + eps*I
    __syncthreads();
    // in-place Gauss-Jordan inversion of Sm (diagonally dominant; no pivoting)
    {
      int c  = tid & 63;
      int rh = tid >> 6;   // 0/1: two rows processed in parallel
      for (int kk = 0; kk < 64; ++kk) {
        if (tid < 64) tv[tid] = Sm[tid*64 + kk];   // capture column kk pre-scale
        __syncthreads();
        float ip = 1.f / tv[kk];
        if (tid < 64) Sm[kk*64 + tid] = (tid == kk) ? ip : Sm[kk*64 + tid] * ip;
        __syncthreads();
        float rk = Sm[kk*64 + c];
        for (int i = rh; i < 64; i += 2) {
          if (i != kk) {
            float f = tv[i];
            Sm[i*64 + c] = (c == kk) ? (-f * ip) : fmaf(-f, rk, Sm[i*64 + c]);
          }
        }
        __syncthreads();
      }
    }
    gemm64<GEMM_STORE>(P, Hs, PH, tid);       // PH = Pp @ H
    __syncthreads();
    gemm64<GEMM_STORE>(PH, Sm, PH, tid);      // Kk = PH @ S^-1 (per-wave in place)
    __syncthreads();
    // innov = z_k - H @ x ; x += Kk @ innov
    if (tid < 64) {
      float acc = z[(((s*K_ + k)*B_ + b)*16 + n)*64 + tid];
      for (int e = 0; e < 64; ++e) acc -= Hs[tid*64 + e] * xv[e];
      iv[tid] = acc;
    }
    __syncthreads();
    if (tid < 64) {
      float acc = xv[tid];
      for (int e = 0; e < 64; ++e) acc += PH[tid*64 + e] * iv[e];
      xv[tid] = acc;
    }
    __syncthreads();
    gemm64<GEMM_SUB>(PH, HP, P, tid);         // P = Pp - Kk @ HP = (I-KH)Pp
    __syncthreads();
  }
  if (tid < 64) x_fin[((s*B_ + b)*16 + n)*64 + tid] = xv[tid];
}

// ---------------------------------------------------------------------------
// Kernel 4a: 3x3 SAME conv (zero pad) on 4x4x64 + ReLU + 1x1 projection.
// grid = S*B, 64 threads.
// ---------------------------------------------------------------------------
__global__ void k_conv(const float* __restrict__ x_fin, const float* __restrict__ Wt1,
                       const float* __restrict__ bt1, const float* __restrict__ Wt2,
                       const float* __restrict__ bt2, float* __restrict__ y_small) {
  int bid = blockIdx.x;          // s*8 + b
  int s = bid >> 3, b = bid & 7;
  int tid = threadIdx.x;         // output channel
  __shared__ float xm[1024];     // [c][n]
  __shared__ float y1[1024];     // [c][n]
  for (int idx = tid; idx < 1024; idx += 64) {
    int cc = idx >> 4, nn = idx & 15;
    xm[idx] = x_fin[((s*B_ + b)*16 + nn)*64 + cc];
  }
  __syncthreads();
  float out[16];
  float bb = bt1[s*64 + tid];
  for (int n = 0; n < 16; ++n) out[n] = bb;
  const float* Wp = Wt1 + (s*64 + tid) * 64 * 9;
  for (int ic = 0; ic < 64; ++ic) {
    float w[9];
    for (int q = 0; q < 9; ++q) w[q] = Wp[ic*9 + q];
    const float* xr = xm + ic * 16;
    for (int i = 0; i < 4; ++i)
      for (int j = 0; j < 4; ++j) {
        float acc = 0.f;
        for (int dy = 0; dy < 3; ++dy) {
          int yy = i + dy - 1;
          if ((unsigned)yy < 4u)
            for (int dx = 0; dx < 3; ++dx) {
              int xx = j + dx - 1;
              if ((unsigned)xx < 4u) acc += w[dy*3 + dx] * xr[yy*4 + xx];
            }
        }
        out[i*4 + j] += acc;
      }
  }
  for (int n = 0; n < 16; ++n) y1[tid*16 + n] = fmaxf(out[n], 0.f);
  __syncthreads();
  const float* W2r = Wt2 + (s*64 + tid) * 64;
  float bb2 = bt2[s*64 + tid];
  for (int n = 0; n < 16; ++n) {
    float acc = bb2;
    for (int cc = 0; cc < 64; ++cc) acc += W2r[cc] * y1[cc*16 + n];
    y_small[((s*B_ + b)*64 + tid)*16 + n] = acc;
  }
}

// ---------------------------------------------------------------------------
// Kernel 4b: bilinear upsample 4x4 -> 96x96, half-pixel, edge clamp.
// One thread per output element of Z[S,B,64,96,96].
// ---------------------------------------------------------------------------
__global__ void k_upsample(const float* __restrict__ y_small, float* __restrict__ Z) {
  int idx = blockIdx.x * 256 + threadIdx.x;       // 18,874,368 total (exact grid)
  int w = idx % 96; int t = idx / 96;
  int h = t % 96; t /= 96;
  int d = t % 64; t /= 64;
  int b = t % 8;  int s = t / 8;
  float sy = (h + 0.5f) * (1.f/24.f) - 0.5f;
  float sx = (w + 0.5f) * (1.f/24.f) - 0.5f;
  float fy = floorf(sy), fx = floorf(sx);
  float wy = sy - fy, wx = sx - fx;
  int i0 = (int)fy, j0 = (int)fx;
  int i1 = i0 + 1, j1 = j0 + 1;
  i0 = min(max(i0, 0), 3); i1 = min(max(i1, 0), 3);
  j0 = min(max(j0, 0), 3); j1 = min(max(j1, 0), 3);
  const float* yp = y_small + ((s*8 + b)*64 + d) * 16;
  float v00 = yp[i0*4 + j0], v01 = yp[i0*4 + j1];
  float v10 = yp[i1*4 + j0], v11 = yp[i1*4 + j1];
  Z[idx] = (1.f - wy) * ((1.f - wx)*v00 + wx*v01) +
           wy         * ((1.f - wx)*v10 + wx*v11);
}

// ---------------------------------------------------------------------------
// Kernel 5: V[s,b,h,w] = mean_c var_k(T) (ddof=1). One thread per pixel.
// ---------------------------------------------------------------------------
__global__ void k_var(const float* __restrict__ T, float* __restrict__ V) {
  int idx = blockIdx.x * 256 + threadIdx.x;       // 294,912 total (exact grid)
  int w = idx % 96; int t = idx / 96;
  int h = t % 96; t /= 96;
  int b = t % 8;  int s = t / 8;
  const float* base = T + (s*32 + b) * CHW_ + h*96 + w;  // k=0,c=0
  float acc = 0.f;
  for (int c = 0; c < 64; ++c) {
    const float* p = base + c * HW_;
    float s1 = 0.f, s2 = 0.f;
    for (int k = 0; k < 4; ++k) {
      float v = p[k * KSTRIDE_];
      s1 += v; s2 += v * v;
    }
    acc += (s2 - s1*s1*0.25f) * (1.f/3.f);
  }
  V[idx] = acc * (1.f/64.f);
}

// ---------------------------------------------------------------------------
extern "C" void kernel_launch(void* const* d_in, const int* in_sizes, int n_in,
                              void* d_out, int out_size, void* d_ws, size_t ws_size,
                              hipStream_t stream) {
  (void)in_sizes; (void)n_in; (void)out_size; (void)ws_size;
  const float* T   = (const float*)d_in[0];
  const float* Q   = (const float*)d_in[1];
  const float* Hm  = (const float*)d_in[2];
  const float* rb  = (const float*)d_in[3];
  const float* W1  = (const float*)d_in[4];
  const float* b1  = (const float*)d_in[5];
  const float* W2  = (const float*)d_in[6];
  const float* b2  = (const float*)d_in[7];
  const float* Wt1 = (const float*)d_in[8];
  const float* bt1 = (const float*)d_in[9];
  const float* Wt2 = (const float*)d_in[10];
  const float* bt2 = (const float*)d_in[11];

  float* ws     = (float*)d_ws;
  float* pooled = ws;            // 6144 : [S,K,B,3,16]
  float* gmax   = ws + 6144;     // 16   : [S,K]
  float* z      = ws + 6160;     // 32768: [S,K,B,N,64]
  float* alpha  = ws + 38928;    // 2048 : [S,K,B,N]
  float* x_fin  = ws + 40976;    // 32768: [S,B,N,64]
  float* y_sm   = ws + 73744;    // 32768: [S,B,64,16]

  float* Z = (float*)d_out;            // [S,B,64,96,96]
  float* V = Z + 18874368;             // [S,B,96,96]

  hipMemsetAsync(gmax, 0, 16 * sizeof(float), stream);
  k_stats<<<2048, 256, 0, stream>>>(T, pooled, gmax);
  k_se<<<128, 64, 0, stream>>>(pooled, gmax, W1, b1, W2, b2, z, alpha);
  k_kalman<<<512, 128, 20672 * sizeof(float), stream>>>(Q, Hm, rb, z, alpha, x_fin);
  k_conv<<<32, 64, 0, stream>>>(x_fin, Wt1, bt1, Wt2, bt2, y_sm);
  k_upsample<<<73728, 256, 0, stream>>>(y_sm, Z);
  k_var<<<1152, 256, 0, stream>>>(T, V);
}